// STEncoder_20718922236131
// MI455X (gfx1250) — compile-verified
//
#include <hip/hip_runtime.h>
#include <math.h>

// ---------------------------------------------------------------------------
// ST-Encoder for MI455X (gfx1250, wave32).
// All dense ops routed through V_WMMA_F32_16X16X4_F32 (fp32-exact tensor path).
// ---------------------------------------------------------------------------

typedef __attribute__((ext_vector_type(2))) float v2f;
typedef __attribute__((ext_vector_type(8))) float v8f;

#define B_ 64
#define T_ 20
#define N_ 100
#define NP_ 112                // adj padded to 112x112 (zero-filled)
#define F_ 64
#define H_ 256
#define BT_ (B_ * T_)          // 1280
#define M_ROWS (B_ * T_ * N_)  // 128000 (divisible by 64)

__device__ __forceinline__ v8f wmma4(v2f a, v2f b, v8f c) {
  return __builtin_amdgcn_wmma_f32_16x16x4_f32(false, a, false, b, (short)0, c,
                                               false, false);
}

// ---------------------- zero-padded adjacency (once) -----------------------
__global__ __launch_bounds__(256) void pad_adj(const float* __restrict__ adj,
                                               float* __restrict__ adjP) {
  const int i = blockIdx.x * 256 + threadIdx.x;
  if (i < NP_ * NP_) {
    const int n = i / NP_, m = i - n * NP_;
    adjP[i] = (n < N_ && m < N_) ? adj[n * N_ + m] : 0.f;
  }
}

// ------------------------------ generic GEMM -------------------------------
// C[M,N] = act(A[M,K] @ W[K,N] + bias) (+ res).  M%64==0, N%64==0, K%16==0.
// Block: 256 threads = 8 waves; block tile 64x64; wave tile 16x32 (2 WMMA/k).
// LDS double-buffered: next K-tile is fetched into registers during compute,
// one barrier per tile.
#define ASP 20  // A LDS stride: mult-of-4 (float4 stores), conflict-free cols
#define BSP 68  // B LDS stride: mult-of-4, conflict-free row reads
template <bool RELU, bool RES>
__global__ __launch_bounds__(256) void gemm_wmma(
    const float* __restrict__ A, const float* __restrict__ W,
    const float* __restrict__ bias, const float* __restrict__ res,
    float* __restrict__ C, int M, int K, int N) {
  __shared__ float As[2][64 * ASP];
  __shared__ float Bs[2][16 * BSP];

  const int tid = threadIdx.x;
  const int wave = tid >> 5;
  const int lane = tid & 31;
  const int ml = lane & 15;  // row/col within 16
  const int hi = lane >> 4;  // half-wave select
  const int wm = wave >> 1;  // 0..3 -> M sub-tile
  const int wn = wave & 1;   // 0..1 -> N sub-tile (32 cols each)
  const int rowBase = blockIdx.y * 64;
  const int colBase = blockIdx.x * 64;

  // staging coords (one float4 per thread per tile)
  const int ar = tid >> 2, ac = (tid & 3) << 2;   // A: 64x16
  const int br = tid >> 4, bc = (tid & 15) << 2;  // B: 16x64
  const float* aPtr = A + (rowBase + ar) * K + ac;
  const float* wPtr = W + br * N + colBase + bc;

  // prologue: stage tile 0
  float4 aReg = *(const float4*)aPtr;
  float4 bReg = *(const float4*)wPtr;
  *(float4*)&As[0][ar * ASP + ac] = aReg;
  *(float4*)&Bs[0][br * BSP + bc] = bReg;
  __syncthreads();

  v8f acc0 = {}, acc1 = {};
  const int nT = K >> 4;
  for (int t = 0; t < nT; ++t) {
    const int cur = t & 1;
    if (t + 1 < nT) {  // prefetch next tile into registers (overlaps WMMAs)
      aReg = *(const float4*)(aPtr + (t + 1) * 16);
      bReg = *(const float4*)(wPtr + (t + 1) * 16 * N);
    }
    const float* Asc = As[cur];
    const float* Bsc = Bs[cur];
#pragma unroll
    for (int kk = 0; kk < 16; kk += 4) {
      const int ka = kk + 2 * hi;  // K pair {0,1} low lanes, {2,3} high lanes
      const v2f a = *(const v2f*)&Asc[(wm * 16 + ml) * ASP + ka];  // b64
      v2f b0, b1;
      b0.x = Bsc[ka * BSP + wn * 32 + ml];
      b0.y = Bsc[(ka + 1) * BSP + wn * 32 + ml];
      b1.x = Bsc[ka * BSP + wn * 32 + 16 + ml];
      b1.y = Bsc[(ka + 1) * BSP + wn * 32 + 16 + ml];
      acc0 = wmma4(a, b0, acc0);
      acc1 = wmma4(a, b1, acc1);
    }
    if (t + 1 < nT) {
      *(float4*)&As[cur ^ 1][ar * ASP + ac] = aReg;
      *(float4*)&Bs[cur ^ 1][br * BSP + bc] = bReg;
      __syncthreads();  // readers of buf[cur] done; buf[cur^1] now visible
    }
  }

  const int col0 = colBase + wn * 32 + ml;
  const float bv0 = bias[col0];
  const float bv1 = bias[col0 + 16];
#pragma unroll
  for (int r = 0; r < 8; ++r) {
    const int row = rowBase + wm * 16 + r + 8 * hi;  // C frag: row = r + 8*hi
    float v0 = acc0[r] + bv0;
    float v1 = acc1[r] + bv1;
    if (RELU) {
      v0 = v0 > 0.f ? v0 : 0.f;
      v1 = v1 > 0.f ? v1 : 0.f;
    }
    if (RES) {
      v0 += res[row * N + col0];
      v1 += res[row * N + col0 + 16];
    }
    C[row * N + col0] = v0;
    C[row * N + col0 + 16] = v1;
  }
}

// -------------------- GCN message passing: out = adjP @ h ------------------
// Per (b,t) slice: out[n,c] = sum_m adj[n,m] * h[m,c].  adjP zero-padded, so
// no load guards; only the store is masked to n<100.  Wave tile 16x32.
__global__ __launch_bounds__(256) void mp_wmma(const float* __restrict__ adjP,
                                               const float* __restrict__ hin,
                                               float* __restrict__ gout,
                                               int C) {
  const int tid = threadIdx.x;
  const int wave = tid >> 5;
  const int lane = tid & 31;
  const int ml = lane & 15;
  const int hi = lane >> 4;
  const int cT = C >> 5;  // 32-col tiles
  const int tile = blockIdx.x * 8 + wave;
  if (tile >= 7 * cT) return;  // wave-uniform exit; EXEC stays all-1s
  const int nt = tile / cT;
  const int ct = tile - nt * cT;

  const float* src = hin + blockIdx.y * (N_ * C);
  float* dst = gout + blockIdx.y * (N_ * C);
  const int nrow = nt * 16 + ml;  // <= 111, in-bounds of adjP
  const int col0 = ct * 32 + ml;
  const float* arow = adjP + nrow * NP_;

  v8f acc0 = {}, acc1 = {};
#pragma unroll 5
  for (int k0 = 0; k0 < 100; k0 += 4) {
    const int ka = k0 + 2 * hi;              // even, <= 98
    const v2f a = *(const v2f*)(arow + ka);  // global_load_b64
    const float* s0 = src + ka * C + col0;
    v2f b0, b1;
    b0.x = s0[0];
    b0.y = s0[C];
    b1.x = s0[16];
    b1.y = s0[C + 16];
    acc0 = wmma4(a, b0, acc0);
    acc1 = wmma4(a, b1, acc1);
  }
#pragma unroll
  for (int r = 0; r < 8; ++r) {
    const int row = nt * 16 + r + 8 * hi;
    if (row < N_) {
      dst[row * C + col0] = acc0[r];
      dst[row * C + col0 + 16] = acc1[r];
    }
  }
}

// -------------------- attention scores + softmax (per b,n) -----------------
#define HP_ (H_ + 1)  // LDS row pad -> conflict-free strided dot products
__global__ __launch_bounds__(256) void attn_scores(const float* __restrict__ q,
                                                   const float* __restrict__ k,
                                                   float* __restrict__ S) {
  __shared__ float sq[T_ * HP_];
  __shared__ float sk[T_ * HP_];
  __shared__ float ss[T_ * T_];
  const int bn = blockIdx.x;
  const int b = bn / N_, n = bn - b * N_;
  const int tid = threadIdx.x;

  for (int i = tid; i < T_ * H_; i += 256) {
    const int t = i >> 8, h = i & (H_ - 1);
    const int g = ((b * T_ + t) * N_ + n) * H_ + h;
    sq[t * HP_ + h] = q[g];
    sk[t * HP_ + h] = k[g];
  }
  __syncthreads();

  for (int e = tid; e < T_ * T_; e += 256) {
    const int t = e / T_, s = e - t * T_;
    const float* qt = sq + t * HP_;
    const float* ks = sk + s * HP_;
    float acc = 0.f;
#pragma unroll 8
    for (int h = 0; h < H_; ++h) acc += qt[h] * ks[h];
    ss[e] = acc * 0.0625f;  // 1/sqrt(256)
  }
  __syncthreads();

  if (tid < T_) {
    const int t = tid;
    float mx = -1e30f;
    for (int s = 0; s < T_; ++s) mx = fmaxf(mx, ss[t * T_ + s]);
    float ex[T_], sum = 0.f;
    for (int s = 0; s < T_; ++s) {
      ex[s] = __expf(ss[t * T_ + s] - mx);
      sum += ex[s];
    }
    const float inv = 1.f / sum;
    float* Sp = S + bn * (T_ * T_) + t * T_;
    for (int s = 0; s < T_; ++s) Sp[s] = ex[s] * inv;
  }
}

// ------------------------- attn @ V (per b,n) ------------------------------
__global__ __launch_bounds__(256) void attn_av(const float* __restrict__ S,
                                               const float* __restrict__ v,
                                               float* __restrict__ out) {
  __shared__ float sv[T_ * H_];
  __shared__ float ssc[T_ * T_];
  const int bn = blockIdx.x;
  const int b = bn / N_, n = bn - b * N_;
  const int tid = threadIdx.x;

  for (int i = tid; i < T_ * H_; i += 256) {
    const int t = i >> 8, h = i & (H_ - 1);
    sv[i] = v[((b * T_ + t) * N_ + n) * H_ + h];
  }
  for (int i = tid; i < T_ * T_; i += 256) ssc[i] = S[bn * (T_ * T_) + i];
  __syncthreads();

  const int h = tid;  // 256 threads == H columns
  for (int t = 0; t < T_; ++t) {
    float acc = 0.f;
#pragma unroll
    for (int s = 0; s < T_; ++s) acc += ssc[t * T_ + s] * sv[s * H_ + h];
    out[((b * T_ + t) * N_ + n) * H_ + h] = acc;
  }
}

// ------------------------------- LayerNorm ---------------------------------
// One wave32 per row of 256; 8 values/lane; shfl_xor tree reductions.
__global__ __launch_bounds__(256) void layernorm(const float* __restrict__ a,
                                                 const float* __restrict__ gamma,
                                                 const float* __restrict__ beta,
                                                 float* __restrict__ out,
                                                 int M) {
  const int wave = threadIdx.x >> 5;
  const int lane = threadIdx.x & 31;
  const int row = blockIdx.x * 8 + wave;
  if (row >= M) return;
  const float* ar = a + row * H_;

  float vals[8], sum = 0.f;
#pragma unroll
  for (int i = 0; i < 8; ++i) {
    vals[i] = ar[i * 32 + lane];
    sum += vals[i];
  }
#pragma unroll
  for (int off = 16; off > 0; off >>= 1) sum += __shfl_xor(sum, off, 32);
  const float mu = sum * (1.f / H_);

  float vs = 0.f;
#pragma unroll
  for (int i = 0; i < 8; ++i) {
    const float d = vals[i] - mu;
    vs += d * d;
  }
#pragma unroll
  for (int off = 16; off > 0; off >>= 1) vs += __shfl_xor(vs, off, 32);
  const float inv = rsqrtf(vs * (1.f / H_) + 1e-5f);

  float* orow = out + row * H_;
#pragma unroll
  for (int i = 0; i < 8; ++i) {
    const int c = i * 32 + lane;
    orow[c] = (vals[i] - mu) * inv * gamma[c] + beta[c];
  }
}

// ------------------------------- launcher ----------------------------------
extern "C" void kernel_launch(void* const* d_in, const int* in_sizes, int n_in,
                              void* d_out, int out_size, void* d_ws,
                              size_t ws_size, hipStream_t stream) {
  const float* x = (const float*)d_in[0];
  const float* adj = (const float*)d_in[1];
  const float* W0 = (const float*)d_in[2];
  const float* b0 = (const float*)d_in[3];
  const float* W1 = (const float*)d_in[4];
  const float* b1 = (const float*)d_in[5];
  const float* W2 = (const float*)d_in[6];
  const float* b2 = (const float*)d_in[7];
  const float* Wq = (const float*)d_in[8];
  const float* bq = (const float*)d_in[9];
  const float* Wk = (const float*)d_in[10];
  const float* bk = (const float*)d_in[11];
  const float* Wv = (const float*)d_in[12];
  const float* bv = (const float*)d_in[13];
  const float* Wp = (const float*)d_in[14];
  const float* bp = (const float*)d_in[15];
  const float* gamma = (const float*)d_in[16];
  const float* beta = (const float*)d_in[17];
  float* out = (float*)d_out;

  // Workspace: A0 (M x 64, reused for scores) + 3 ping-pong (M x 256) + adjP.
  // Total = 426,034,176 bytes (~406 MiB).
  float* A0 = (float*)d_ws;
  float* Bu1 = A0 + (size_t)M_ROWS * F_;
  float* Bu2 = Bu1 + (size_t)M_ROWS * H_;
  float* Bu3 = Bu2 + (size_t)M_ROWS * H_;
  float* adjP = Bu3 + (size_t)M_ROWS * H_;

  const dim3 blk(256);
  const dim3 gG(H_ / 64, M_ROWS / 64);  // dense GEMM grid (4, 2000)
  const dim3 gMP64((7 * (F_ / 32) + 7) / 8, BT_);   // (2, 1280)
  const dim3 gMP256((7 * (H_ / 32) + 7) / 8, BT_);  // (7, 1280)
  const dim3 gAttn(B_ * N_);

  pad_adj<<<dim3((NP_ * NP_ + 255) / 256), blk, 0, stream>>>(adj, adjP);

  // GCN layer 0: g0 = adj@x ; h0 = relu(g0@W0 + b0)            (no residual)
  mp_wmma<<<gMP64, blk, 0, stream>>>(adjP, x, A0, F_);
  gemm_wmma<true, false><<<gG, blk, 0, stream>>>(A0, W0, b0, nullptr, Bu1,
                                                 M_ROWS, F_, H_);
  // GCN layer 1: h1 = relu((adj@h0)@W1 + b1) + h0
  mp_wmma<<<gMP256, blk, 0, stream>>>(adjP, Bu1, Bu2, H_);
  gemm_wmma<true, true><<<gG, blk, 0, stream>>>(Bu2, W1, b1, Bu1, Bu3, M_ROWS,
                                                H_, H_);
  // GCN layer 2: h2 = relu((adj@h1)@W2 + b2) + h1
  mp_wmma<<<gMP256, blk, 0, stream>>>(adjP, Bu3, Bu1, H_);
  gemm_wmma<true, true><<<gG, blk, 0, stream>>>(Bu1, W2, b2, Bu3, Bu2, M_ROWS,
                                                H_, H_);
  // q = h2@Wq + bq ; k = h2@Wk + bk
  gemm_wmma<false, false><<<gG, blk, 0, stream>>>(Bu2, Wq, bq, nullptr, Bu1,
                                                  M_ROWS, H_, H_);
  gemm_wmma<false, false><<<gG, blk, 0, stream>>>(Bu2, Wk, bk, nullptr, Bu3,
                                                  M_ROWS, H_, H_);
  // scores = softmax(q k^T / 16)  -> A0 (10.2 MB, fits in the 32.8 MB slot)
  attn_scores<<<gAttn, blk, 0, stream>>>(Bu1, Bu3, A0);
  // v = h2@Wv + bv  (overwrites q, already consumed)
  gemm_wmma<false, false><<<gG, blk, 0, stream>>>(Bu2, Wv, bv, nullptr, Bu1,
                                                  M_ROWS, H_, H_);
  // attn_out = scores @ v
  attn_av<<<gAttn, blk, 0, stream>>>(A0, Bu1, Bu3);
  // a = attn_out@Wp + bp
  gemm_wmma<false, false><<<gG, blk, 0, stream>>>(Bu3, Wp, bp, nullptr, Bu2,
                                                  M_ROWS, H_, H_);
  // LayerNorm -> d_out
  layernorm<<<dim3(M_ROWS / 8), blk, 0, stream>>>(Bu2, gamma, beta, out,
                                                  M_ROWS);
}